// EdgeConditionedGATLayer_39994735460483
// MI455X (gfx1250) — compile-verified
//
#include <hip/hip_runtime.h>
#include <math.h>

// ---------------------------------------------------------------------------
// EdgeConditionedGAT layer for MI455X (gfx1250, wave32).
// fp32 WMMA (v_wmma_f32_16x16x4_f32) for the node transform, the edge-feature
// transform AND the gate-MLP hidden layer (fused on shared A fragments);
// float atomics for segment softmax / sums; wave shuffles for reductions.
// ---------------------------------------------------------------------------

typedef __attribute__((ext_vector_type(2))) float v2f;
typedef __attribute__((ext_vector_type(8))) float v8f;

#define HEADS 4
#define DIM 64
#define EDIM 16
#define GHID 32          // gate MLP hidden width
#define NEG_SLOPE 0.2f
#define LN_EPS 1e-5f

// IEEE-ordered float atomic max via signed-max / unsigned-min trick.
__device__ __forceinline__ void atomicMaxF(float* addr, float v) {
  if (v >= 0.f)
    atomicMax((int*)addr, __float_as_int(v));
  else
    atomicMin((unsigned int*)addr, (unsigned int)__float_as_int(v));
}

// ---------------------------------------------------------------------------
// Kernel 0: initialize workspace accumulators.
// ---------------------------------------------------------------------------
__global__ void init_ws(float* __restrict__ accum, float* __restrict__ amax,
                        float* __restrict__ denom, float* __restrict__ gsum,
                        float* __restrict__ deg, int N) {
  int i = blockIdx.x * blockDim.x + threadIdx.x;
  if (i < N * DIM) accum[i] = 0.f;
  if (i < N * HEADS) { amax[i] = -INFINITY; denom[i] = 0.f; }
  if (i < N) { gsum[i] = 0.f; deg[i] = 0.f; }
}

// ---------------------------------------------------------------------------
// Kernel 1: node transform. One wave per 16-node tile.
// xl = x @ Wl + bl ; xr = x @ Wr + br   (f32 WMMA, K=64 in 16 steps of 4)
// A frag (16x4 f32): lane<16 -> row m=lane&15, k=kk+{0,1}; lane>=16 -> k=kk+{2,3}
// C/D (16x16 f32): VGPR r, lanes 0-15 -> M=r, lanes 16-31 -> M=r+8; N=lane&15.
// ---------------------------------------------------------------------------
__global__ void node_xform(const float* __restrict__ x,
                           const float* __restrict__ Wl, const float* __restrict__ bl,
                           const float* __restrict__ Wr, const float* __restrict__ br,
                           float* __restrict__ xl, float* __restrict__ xr,
                           int nTiles) {
  int wave = (blockIdx.x * blockDim.x + threadIdx.x) >> 5;
  int lane = threadIdx.x & 31;
  if (wave >= nTiles) return;                 // wave-uniform; EXEC stays all-1s
  int row0 = wave << 4;
  int n15 = lane & 15;
  int kh  = (lane >> 4) << 1;                 // 0 or 2
  const float* xrow = x + (size_t)(row0 + n15) * DIM;

  v8f accL[4] = {};
  v8f accR[4] = {};

#pragma unroll
  for (int kk = 0; kk < DIM; kk += 4) {
    v2f a;
    a.x = xrow[kk + kh];
    a.y = xrow[kk + kh + 1];
#pragma unroll
    for (int nt = 0; nt < 4; ++nt) {
      int n = nt * 16 + n15;
      v2f bL, bR;
      bL.x = Wl[(kk + kh) * DIM + n];
      bL.y = Wl[(kk + kh + 1) * DIM + n];
      bR.x = Wr[(kk + kh) * DIM + n];
      bR.y = Wr[(kk + kh + 1) * DIM + n];
      accL[nt] = __builtin_amdgcn_wmma_f32_16x16x4_f32(
          false, a, false, bL, (short)0, accL[nt], false, false);
      accR[nt] = __builtin_amdgcn_wmma_f32_16x16x4_f32(
          false, a, false, bR, (short)0, accR[nt], false, false);
    }
  }

  int mo = (lane >> 4) << 3;                  // 0 or 8
#pragma unroll
  for (int nt = 0; nt < 4; ++nt) {
    int col = nt * 16 + n15;
    float bLv = bl[col], bRv = br[col];
#pragma unroll
    for (int r = 0; r < 8; ++r) {
      size_t o = (size_t)(row0 + mo + r) * DIM + col;
      xl[o] = accL[nt][r] + bLv;
      xr[o] = accR[nt][r] + bRv;
    }
  }
}

// ---------------------------------------------------------------------------
// Kernel 2: fused edge pass. One wave per 16-edge tile.
//  (a) ee = edge_attr @ We  via WMMA (4 N-tiles)
//  (b) gate hidden = edge_attr @ Wg1 via WMMA (2 N-tiles, SAME A fragments)
// C layout: edge = row (VGPR idx + lane-half), col = nt*16 + (lane&15).
// Then: leaky-ReLU(ee + xl[src] + xr[tgt]) . att  -> alpha[E,H], amax atomicMax
//       sigmoid(Wg2 . silu(hidden + bg1) + bg2)   -> gsum/deg atomicAdd
// Both contractions use the 16-lane-half shfl_xor reduction tree.
// ---------------------------------------------------------------------------
__global__ void edge_alpha_gate(const float* __restrict__ ea,
                                const int* __restrict__ srcI, const int* __restrict__ tgtI,
                                const float* __restrict__ We, const float* __restrict__ att,
                                const float* __restrict__ Wg1, const float* __restrict__ bg1,
                                const float* __restrict__ Wg2, const float* __restrict__ bg2,
                                const float* __restrict__ xl, const float* __restrict__ xr,
                                float* __restrict__ alphaW, float* __restrict__ amax,
                                float* __restrict__ gsum, float* __restrict__ deg,
                                int eTiles) {
  int wave = (blockIdx.x * blockDim.x + threadIdx.x) >> 5;
  int lane = threadIdx.x & 31;
  if (wave >= eTiles) return;                 // wave-uniform
  int e0  = wave << 4;
  int n15 = lane & 15;
  int kh  = (lane >> 4) << 1;
  const float* earow = ea + (size_t)(e0 + n15) * EDIM;

  v8f acc[4]  = {};                           // ee accumulators (4 x 16 cols)
  v8f gacc[2] = {};                           // gate hidden accumulators (2 x 16)
#pragma unroll
  for (int kk = 0; kk < EDIM; kk += 4) {
    v2f a;
    a.x = earow[kk + kh];
    a.y = earow[kk + kh + 1];
#pragma unroll
    for (int nt = 0; nt < 4; ++nt) {
      int n = nt * 16 + n15;
      v2f b;
      b.x = We[(kk + kh) * DIM + n];
      b.y = We[(kk + kh + 1) * DIM + n];
      acc[nt] = __builtin_amdgcn_wmma_f32_16x16x4_f32(
          false, a, false, b, (short)0, acc[nt], false, false);
    }
#pragma unroll
    for (int g = 0; g < 2; ++g) {
      int n = g * 16 + n15;
      v2f b;
      b.x = Wg1[(kk + kh) * GHID + n];
      b.y = Wg1[(kk + kh + 1) * GHID + n];
      gacc[g] = __builtin_amdgcn_wmma_f32_16x16x4_f32(
          false, a, false, b, (short)0, gacc[g], false, false);
    }
  }

  // per-lane constants for the contractions
  float attv[4];
#pragma unroll
  for (int nt = 0; nt < 4; ++nt) attv[nt] = att[nt * 16 + n15];
  float bg1v[2] = { bg1[n15], bg1[16 + n15] };
  float wg2v[2] = { Wg2[n15], Wg2[16 + n15] };
  float bg2v = bg2[0];

  int mo = (lane >> 4) << 3;
#pragma unroll
  for (int r = 0; r < 8; ++r) {
    int e = e0 + mo + r;
    int s = srcI[e];
    int t = tgtI[e];

    // ---- attention logits (4 heads) ----
#pragma unroll
    for (int nt = 0; nt < 4; ++nt) {
      int c = nt * 16 + n15;
      float mv = acc[nt][r] + xl[(size_t)s * DIM + c] + xr[(size_t)t * DIM + c];
      mv = mv >= 0.f ? mv : NEG_SLOPE * mv;   // leaky relu
      float p = mv * attv[nt];
      p += __shfl_xor(p, 1, 32);
      p += __shfl_xor(p, 2, 32);
      p += __shfl_xor(p, 4, 32);
      p += __shfl_xor(p, 8, 32);
      if (n15 == 0) {
        alphaW[(size_t)e * HEADS + nt] = p;
        atomicMaxF(&amax[(size_t)t * HEADS + nt], p);
      }
    }

    // ---- edge gate: sigmoid(Wg2 . silu(hidden)) ----
    float gp = 0.f;
#pragma unroll
    for (int g = 0; g < 2; ++g) {
      float h = gacc[g][r] + bg1v[g];
      float sl = h / (1.f + expf(-h));        // silu
      gp += sl * wg2v[g];
    }
    gp += __shfl_xor(gp, 1, 32);
    gp += __shfl_xor(gp, 2, 32);
    gp += __shfl_xor(gp, 4, 32);
    gp += __shfl_xor(gp, 8, 32);
    if (n15 == 0) {
      float gate = 1.f / (1.f + expf(-(gp + bg2v)));
      atomicAdd(&gsum[t], gate);
      atomicAdd(&deg[t], 1.f);
    }
  }
}

// ---------------------------------------------------------------------------
// Kernel 3: softmax-weighted scatter. One thread per (edge, channel).
// a = exp(alpha - amax[tgt]); accum[tgt,c] += a * xl[src,c]; denom[tgt,h] += a.
// Division by denom deferred to the node pass (saves a full edge pass).
// ---------------------------------------------------------------------------
__global__ void edge_scatter(const int* __restrict__ srcI, const int* __restrict__ tgtI,
                             const float* __restrict__ alphaW, const float* __restrict__ amax,
                             const float* __restrict__ xl,
                             float* __restrict__ accum, float* __restrict__ denom, int E) {
  int idx = blockIdx.x * blockDim.x + threadIdx.x;     // E*64 = 102.4M fits int
  if (idx >= E * DIM) return;
  int e = idx >> 6;
  int c = idx & 63;
  int h = c >> 4;
  int s = srcI[e];
  int t = tgtI[e];
  float a = expf(alphaW[(size_t)e * HEADS + h] - amax[(size_t)t * HEADS + h]);
  atomicAdd(&accum[(size_t)t * DIM + c], a * xl[(size_t)s * DIM + c]);
  if ((c & 15) == 0) atomicAdd(&denom[(size_t)t * HEADS + h], a);
}

// ---------------------------------------------------------------------------
// Kernel 4: per-node finalize. One wave per node, 2 channels per lane.
// out = silu(LN((accum/denom + bias) * mean_gate)) + x
// ---------------------------------------------------------------------------
__global__ void node_finalize(const float* __restrict__ accum, const float* __restrict__ denom,
                              const float* __restrict__ gsum, const float* __restrict__ deg,
                              const float* __restrict__ conv_bias,
                              const float* __restrict__ gamma, const float* __restrict__ beta,
                              const float* __restrict__ x, float* __restrict__ out, int N) {
  int node = (blockIdx.x * blockDim.x + threadIdx.x) >> 5;
  int lane = threadIdx.x & 31;
  if (node >= N) return;
  float mg = gsum[node] / fmaxf(deg[node], 1.f);
  float vv[2];
  float sum = 0.f, sumsq = 0.f;
#pragma unroll
  for (int q = 0; q < 2; ++q) {
    int c = lane * 2 + q;
    int h = c >> 4;
    float dn = denom[(size_t)node * HEADS + h];
    float o = dn > 0.f ? accum[(size_t)node * DIM + c] / dn : 0.f;  // empty segment -> 0
    o = (o + conv_bias[c]) * mg;
    vv[q] = o;
    sum += o;
    sumsq += o * o;
  }
#pragma unroll
  for (int m = 1; m < 32; m <<= 1) {
    sum   += __shfl_xor(sum, m, 32);
    sumsq += __shfl_xor(sumsq, m, 32);
  }
  float mu  = sum * (1.f / DIM);
  float var = sumsq * (1.f / DIM) - mu * mu;
  float inv = rsqrtf(var + LN_EPS);
#pragma unroll
  for (int q = 0; q < 2; ++q) {
    int c = lane * 2 + q;
    float y = (vv[q] - mu) * inv * gamma[c] + beta[c];
    y = y / (1.f + expf(-y));                 // silu
    out[(size_t)node * DIM + c] = y + x[(size_t)node * DIM + c];
  }
}

// ---------------------------------------------------------------------------
extern "C" void kernel_launch(void* const* d_in, const int* in_sizes, int n_in,
                              void* d_out, int out_size, void* d_ws, size_t ws_size,
                              hipStream_t stream) {
  const float* x     = (const float*)d_in[0];
  const int*   eidx  = (const int*)d_in[1];
  const float* ea    = (const float*)d_in[2];
  const float* Wl    = (const float*)d_in[3];
  const float* bl    = (const float*)d_in[4];
  const float* Wr    = (const float*)d_in[5];
  const float* br    = (const float*)d_in[6];
  const float* We    = (const float*)d_in[7];
  const float* att   = (const float*)d_in[8];
  const float* cb    = (const float*)d_in[9];
  const float* Wg1   = (const float*)d_in[10];
  const float* bg1   = (const float*)d_in[11];
  const float* Wg2   = (const float*)d_in[12];
  const float* bg2   = (const float*)d_in[13];
  const float* gamma = (const float*)d_in[14];
  const float* beta  = (const float*)d_in[15];
  float* out = (float*)d_out;

  int N = in_sizes[0] / DIM;    // 100000
  int E = in_sizes[2] / EDIM;   // 1600000
  const int* srcI = eidx;       // edge_index[0,:]
  const int* tgtI = eidx + E;   // edge_index[1,:]

  // Workspace layout (floats): ~106.4 MB total
  float* ws     = (float*)d_ws;
  float* xl     = ws;                               // N*64
  float* xr     = xl + (size_t)N * DIM;             // N*64
  float* accum  = xr + (size_t)N * DIM;             // N*64
  float* alphaW = accum + (size_t)N * DIM;          // E*4
  float* amax   = alphaW + (size_t)E * HEADS;       // N*4
  float* denom  = amax + (size_t)N * HEADS;         // N*4
  float* gsum   = denom + (size_t)N * HEADS;        // N
  float* deg    = gsum + N;                         // N

  {
    int total = N * DIM;
    init_ws<<<(total + 255) / 256, 256, 0, stream>>>(accum, amax, denom, gsum, deg, N);
  }
  {
    int tiles = (N + 15) / 16;                      // 6250
    int wpb = 4;                                    // 128 threads = 4 waves
    node_xform<<<(tiles + wpb - 1) / wpb, wpb * 32, 0, stream>>>(
        x, Wl, bl, Wr, br, xl, xr, tiles);
  }
  {
    int tiles = (E + 15) / 16;                      // 100000
    int wpb = 8;                                    // 256 threads = 8 waves
    edge_alpha_gate<<<(tiles + wpb - 1) / wpb, wpb * 32, 0, stream>>>(
        ea, srcI, tgtI, We, att, Wg1, bg1, Wg2, bg2, xl, xr,
        alphaW, amax, gsum, deg, tiles);
  }
  {
    long long total = (long long)E * DIM;           // 102.4M threads
    int blocks = (int)((total + 255) / 256);
    edge_scatter<<<blocks, 256, 0, stream>>>(srcI, tgtI, alphaW, amax, xl, accum, denom, E);
  }
  {
    int wpb = 8;
    node_finalize<<<(N + wpb - 1) / wpb, wpb * 32, 0, stream>>>(
        accum, denom, gsum, deg, cb, gamma, beta, x, out, N);
  }
}